// HomConv_22900765622290
// MI455X (gfx1250) — compile-verified
//
#include <hip/hip_runtime.h>

// HomConv over path P4 unrolled:
//   deg = indegree(edge_dst)
//   h2  = relu(w*relu(w+b)*deg + b)
//   s2  = segment_sum(h2[src] by dst)
//   h1  = relu(w*s2 + b)
//   s1  = segment_sum(h1[src] by dst)
//   out = sum(relu(w*s1 + b))
// Memory-bound: ~320MB edge streaming (floor ~14us @ 23.3TB/s); node arrays
// (4MB) are L2-resident so gathers + f32 scatter atomics resolve in L2.
// Final reduction uses V_WMMA_F32_16X16X4_F32 (B=ones -> exact f32 row sums).

typedef __attribute__((ext_vector_type(2))) float v2f;
typedef __attribute__((ext_vector_type(8))) float v8f;

__device__ __forceinline__ float relu_f(float x) { return x > 0.0f ? x : 0.0f; }

// Zero deg (int), s2 (float), and the scalar output.
__global__ void k_init(int* deg, float* s2, float* out, const int* np) {
  const int n = *np;
  int i = blockIdx.x * blockDim.x + threadIdx.x;
  int stride = gridDim.x * blockDim.x;
  for (int a = i; a < n; a += stride) { deg[a] = 0; s2[a] = 0.0f; }
  if (i == 0) out[0] = 0.0f;
}

// deg[dst] += 1 over all edges (int4-vectorized, prefetch ahead).
__global__ void k_degree(const int* __restrict__ dst, long nedge,
                         int* __restrict__ deg) {
  long i0 = blockIdx.x * (long)blockDim.x + threadIdx.x;
  long stride = (long)gridDim.x * blockDim.x;
  long ne4 = nedge >> 2;
  const int4* d4 = (const int4*)dst;
  for (long i = i0; i < ne4; i += stride) {
    __builtin_prefetch(&d4[i + stride], 0, 0);  // global_prefetch_b8
    int4 dd = d4[i];
    atomicAdd(&deg[dd.x], 1);
    atomicAdd(&deg[dd.y], 1);
    atomicAdd(&deg[dd.z], 1);
    atomicAdd(&deg[dd.w], 1);
  }
  for (long i = (ne4 << 2) + i0; i < nedge; i += stride)
    atomicAdd(&deg[dst[i]], 1);
}

// h[a] = relu(w*c3*deg[a] + b); also re-zeroes the deg buffer as float s1.
// NOTE: deg and s1_zero alias the same memory (read-then-write per index),
// so neither is __restrict__.
__global__ void k_h2(const int* deg, float* __restrict__ h, float* s1_zero,
                     const float* wp, const float* bp, const int* np) {
  const int n = *np;
  const float w = *wp, b = *bp;
  const float c3 = relu_f(w + b);
  int i = blockIdx.x * blockDim.x + threadIdx.x;
  int stride = gridDim.x * blockDim.x;
  for (int a = i; a < n; a += stride) {
    float d = (float)deg[a];
    h[a] = relu_f(w * c3 * d + b);
    s1_zero[a] = 0.0f;
  }
}

// h[a] = relu(w*s2[a] + b)
__global__ void k_h1(const float* __restrict__ s2, float* __restrict__ h,
                     const float* wp, const float* bp, const int* np) {
  const int n = *np;
  const float w = *wp, b = *bp;
  int i = blockIdx.x * blockDim.x + threadIdx.x;
  int stride = gridDim.x * blockDim.x;
  for (int a = i; a < n; a += stride) h[a] = relu_f(w * s2[a] + b);
}

// s[dst] += h[src] over all edges (float atomics resolve in L2; node arrays
// are L2-resident).
__global__ void k_scatter(const int* __restrict__ src, const int* __restrict__ dst,
                          long nedge, const float* __restrict__ h,
                          float* __restrict__ s) {
  long i0 = blockIdx.x * (long)blockDim.x + threadIdx.x;
  long stride = (long)gridDim.x * blockDim.x;
  long ne4 = nedge >> 2;
  const int4* s4 = (const int4*)src;
  const int4* d4 = (const int4*)dst;
  for (long i = i0; i < ne4; i += stride) {
    __builtin_prefetch(&s4[i + stride], 0, 0);
    __builtin_prefetch(&d4[i + stride], 0, 0);
    int4 ss = s4[i];
    int4 dd = d4[i];
    atomicAdd(&s[dd.x], h[ss.x]);
    atomicAdd(&s[dd.y], h[ss.y]);
    atomicAdd(&s[dd.z], h[ss.z]);
    atomicAdd(&s[dd.w], h[ss.w]);
  }
  for (long i = (ne4 << 2) + i0; i < nedge; i += stride)
    atomicAdd(&s[dst[i]], h[src[i]]);
}

// out += sum(relu(w*s1 + b)).  Per-wave reduction via WMMA f32:
// A(16x4) holds the 32 lane-partials (K=0 lanes 0-15, K=2 lanes 16-31, rest 0),
// B = ones(4x16)  =>  D[m,n] = rowsum_m = acc_m + acc_{m+16} in exact f32.
// Lane 0 holds rowsums 0..7, lane 16 holds rowsums 8..15.
__global__ void k_reduce(const float* __restrict__ s1, const float* wp,
                         const float* bp, const int* np, float* out) {
  const int n = *np;
  const float w = *wp, b = *bp;
  int i = blockIdx.x * blockDim.x + threadIdx.x;
  int stride = gridDim.x * blockDim.x;
  float acc = 0.0f;
  for (int a = i; a < n; a += stride) acc += relu_f(w * s1[a] + b);

  v2f A = {acc, 0.0f};
  v2f B = {1.0f, 1.0f};
  v8f C = {};
  v8f D = __builtin_amdgcn_wmma_f32_16x16x4_f32(
      /*neg_a=*/false, A, /*neg_b=*/false, B,
      /*c_mod=*/(short)0, C, /*reuse_a=*/false, /*reuse_b=*/false);
  float lsum = D[0] + D[1] + D[2] + D[3] + D[4] + D[5] + D[6] + D[7];
  float tot = lsum + __shfl_down(lsum, 16, 32);  // lane0: full wave sum
  if ((threadIdx.x & 31) == 0) atomicAdd(out, tot);
}

extern "C" void kernel_launch(void* const* d_in, const int* in_sizes, int n_in,
                              void* d_out, int out_size, void* d_ws, size_t ws_size,
                              hipStream_t stream) {
  const int*   edge_src = (const int*)d_in[0];
  const int*   edge_dst = (const int*)d_in[1];
  const int*   np       = (const int*)d_in[2];   // device scalar n
  const float* wp       = (const float*)d_in[3];
  const float* bp       = (const float*)d_in[4];
  float*       out      = (float*)d_out;
  const long   nedge    = (long)in_sizes[0];

  // Carve 3 node-sized buffers from ws:
  //   buf0: deg (int) -> reused as s1 (float)
  //   buf1: s2 (float)
  //   buf2: h (float), reused for h2 then h1
  size_t cap = (ws_size / 3) & ~(size_t)255;
  char* base = (char*)d_ws;
  int*   buf0i = (int*)(base);
  float* buf0f = (float*)(base);
  float* buf1  = (float*)(base + cap);
  float* buf2  = (float*)(base + 2 * cap);

  const int T = 256;
  long ne4 = nedge >> 2;
  long eb = (ne4 + T - 1) / T;
  int eblocks = (int)(eb < 1 ? 1 : (eb > 2048 ? 2048 : eb));
  const int nblocks = 1024;  // node kernels grid-stride over device n
  const int rblocks = 256;

  k_init<<<nblocks, T, 0, stream>>>(buf0i, buf1, out, np);
  k_degree<<<eblocks, T, 0, stream>>>(edge_dst, nedge, buf0i);
  k_h2<<<nblocks, T, 0, stream>>>(buf0i, buf2, buf0f, wp, bp, np);
  k_scatter<<<eblocks, T, 0, stream>>>(edge_src, edge_dst, nedge, buf2, buf1);
  k_h1<<<nblocks, T, 0, stream>>>(buf1, buf2, wp, bp, np);
  k_scatter<<<eblocks, T, 0, stream>>>(edge_src, edge_dst, nedge, buf2, buf0f);
  k_reduce<<<rblocks, T, 0, stream>>>(buf0f, wp, bp, np, out);
}